// CaceA_79096117723650
// MI455X (gfx1250) — compile-verified
//
#include <hip/hip_runtime.h>
#include <hip/hip_bf16.h>

typedef __attribute__((ext_vector_type(2))) float v2f;
typedef __attribute__((ext_vector_type(8))) float v8f;

#define NABc   4
#define NANGc  10
#define NVALs  320      // NRBF(8) * NANG(10) * NAB(4) per node
#define CUTc   4.0f

__device__ __forceinline__ int z_to_idx(int z) {
    // ZS = {1, 6, 7, 8, 9}
    return (z == 1) ? 0 : (z == 6) ? 1 : (z == 7) ? 2 : (z == 8) ? 3 : 4;
}

__global__ void k_zero(float* __restrict__ B, int n) {
    int i = blockIdx.x * blockDim.x + threadIdx.x;
    if (i < n) B[i] = 0.0f;
}

__global__ void k_emb(const int* __restrict__ zn,
                      const float* __restrict__ Ws, const float* __restrict__ Wr,
                      float* __restrict__ es, float* __restrict__ er, int N) {
    int n = blockIdx.x * blockDim.x + threadIdx.x;
    if (n >= N) return;
    int zi = z_to_idx(zn[n]);
#pragma unroll
    for (int a = 0; a < NABc; ++a) {
        es[n * NABc + a] = Ws[zi * NABc + a];
        er[n * NABc + a] = Wr[zi * NABc + a];
    }
}

// One wave32 per edge; 320 scatter values = 32 lanes x 10 atomics.
// Edge index is forced into an SGPR so all per-edge scalars load scalar-side.
__global__ void k_edge(const int* __restrict__ eidx, const float* __restrict__ dij,
                       const float* __restrict__ uij, const float* __restrict__ widths,
                       const float* __restrict__ es, float* __restrict__ B, int E) {
    int lane = threadIdx.x & 31;
    int e = __builtin_amdgcn_readfirstlane(blockIdx.x * 8 + (threadIdx.x >> 5));
    if (e >= E) return;                 // scalar branch

    int src = eidx[e];
    int dst = eidx[E + e];
    float d = dij[e];
    float x = uij[3 * e + 0], y = uij[3 * e + 1], z = uij[3 * e + 2];

    // polynomial cutoff, p = 5: 1 - 21 x^5 + 35 x^6 - 15 x^7
    float t  = d * (1.0f / CUTc);
    float t2 = t * t;
    float t5 = t2 * t2 * t;
    float env = 1.0f - 21.0f * t5 + 35.0f * t5 * t - 15.0f * t5 * t2;
    env = (d < CUTc) ? env : 0.0f;

    float rad[8];
#pragma unroll
    for (int r = 0; r < 8; ++r) {
        float q = d * __frcp_rn(widths[r]);
        rad[r] = __expf(-0.5f * q * q) * env;
    }
    float ang[10] = {1.0f, x, y, z, x * x, x * y, x * z, y * y, y * z, z * z};

    const float4* es4 = (const float4*)(es);
    float4 ev = es4[src];
    float ea[4] = {ev.x, ev.y, ev.z, ev.w};

    float* Bn = B + dst * NVALs;
#pragma unroll
    for (int k = 0; k < 10; ++k) {
        int v = lane + 32 * k;          // 0..319, layout [r][i][a]
        int r = v / 40;
        int rem = v - r * 40;
        int i = rem >> 2;
        int a = rem & 3;
        atomicAdd(&Bn[v], rad[r] * ang[i] * ea[a]);
    }
}

// Store epilogue for one scalar (i, a) pair. All branches on i are s_cbranch.
template <bool CHK>
__device__ __forceinline__ void emit_stores(float* __restrict__ out, v8f acc,
                                            const float4* __restrict__ ero,
                                            int n0, int half, int bcol,
                                            int i, int a, int N) {
    int a1off = N * 192;
    int a2off = N * 192 * 4;
    int f0 = bcol * 16 + a * 4;
    if (i == 0) {
#pragma unroll
        for (int j = 0; j < 8; ++j) {
            int n = n0 + j + 8 * half;
            if (CHK && n >= N) continue;
            float c = acc[j]; float4 ev = ero[j];
            float4 st = {c * ev.x, c * ev.y, c * ev.z, c * ev.w};
            *(float4*)(out + n * 192 + f0) = st;    // contiguous, 16B-aligned
        }
    } else if (i < 4) {
        int d1 = i - 1;
#pragma unroll
        for (int j = 0; j < 8; ++j) {
            int n = n0 + j + 8 * half;
            if (CHK && n >= N) continue;
            float c = acc[j]; float4 ev = ero[j];
            int base = a1off + (n * 192 + f0) * 3 + d1;
            out[base + 0] = c * ev.x;
            out[base + 3] = c * ev.y;
            out[base + 6] = c * ev.z;
            out[base + 9] = c * ev.w;
        }
    } else {
        // symmetric 3x3 positions: primary p, optional mirror s (all scalar)
        int p = (i == 4) ? 0 : (i == 5) ? 1 : (i == 6) ? 2 : (i == 7) ? 4 : (i == 8) ? 5 : 8;
        bool has2 = (i == 5) || (i == 6) || (i == 8);
        int s = (i == 5) ? 3 : (i == 6) ? 6 : 7;
#pragma unroll
        for (int j = 0; j < 8; ++j) {
            int n = n0 + j + 8 * half;
            if (CHK && n >= N) continue;
            float c = acc[j]; float4 ev = ero[j];
            int base = a2off + (n * 192 + f0) * 9;
            float v0 = c * ev.x, v1 = c * ev.y, v2 = c * ev.z, v3 = c * ev.w;
            out[base + p + 0]  = v0;
            out[base + p + 9]  = v1;
            out[base + p + 18] = v2;
            out[base + p + 27] = v3;
            if (has2) {
                out[base + s + 0]  = v0;
                out[base + s + 9]  = v1;
                out[base + s + 18] = v2;
                out[base + s + 27] = v3;
            }
        }
    }
}

// Node-side contraction, node-tiled: M = 16 nodes, Ncols = 12 (b), K = 8 (r).
// blockIdx.y = ia-group (0..4)  -> ia/i/a/l are SGPR-uniform by construction.
// blockIdx.x*8 + wave = node-tile; tileN forced uniform via readfirstlane so the
// epilogue control flow and store bases are all scalar.
__global__ void k_node_wmma(const float* __restrict__ B, const float* __restrict__ er,
                            const float* __restrict__ Wrad, float* __restrict__ out,
                            int N, int ntiles) {
    int lane = threadIdx.x & 31;
    int wib  = threadIdx.x >> 5;                     // 8 waves per block
    int tileN = __builtin_amdgcn_readfirstlane(blockIdx.x * 8 + wib);
    int g = blockIdx.y;                              // ia group: ia = g*8 .. g*8+7
    if (tileN >= ntiles) return;                     // scalar branch
    int n0 = tileN * 16;
    bool full = (n0 + 16 <= N);                      // scalar

    int half = lane >> 4;               // A/B lanes 16-31 carry K = +2,+3 of each chunk
    int m    = lane & 15;
    int bcol = m;                       // output column b (valid if < 12)
    int bc   = bcol < 12 ? bcol : 11;
    bool bval = bcol < 12;

    int nA = n0 + m;                    // this lane's A-matrix row = node
    bool avalid = nA < N;
    const float* Bn = B + (avalid ? nA : (N - 1)) * NVALs + g * 8;

    // weights for all 3 l-slices, both K-chunks: 12 registers, reused for 8 pairs;
    // invalid columns (bcol >= 12) zeroed once here, not per iteration
    float wv[12];
#pragma unroll
    for (int l = 0; l < 3; ++l) {
#pragma unroll
        for (int c = 0; c < 2; ++c) {
            int r0 = c * 4 + half * 2;
            float u0 = Wrad[l * 96 + r0 * 12 + bc];
            float u1 = Wrad[l * 96 + (r0 + 1) * 12 + bc];
            wv[l * 4 + c * 2 + 0] = bval ? u0 : 0.0f;
            wv[l * 4 + c * 2 + 1] = bval ? u1 : 0.0f;
        }
    }

    // emb_r for this lane's 8 output nodes (D layout: VGPR j -> M = j + 8*half)
    const float4* er4 = (const float4*)er;
    float4 ero[8];
#pragma unroll
    for (int j = 0; j < 8; ++j) {
        int n = n0 + j + 8 * half;
        ero[j] = er4[(n < N) ? n : (N - 1)];
    }

#pragma unroll
    for (int k = 0; k < 8; ++k) {
        int ia = g * 8 + k;             // scalar
        int i  = ia >> 2;
        int a  = ia & 3;
        int l  = (i == 0) ? 0 : (i < 4) ? 1 : 2;   // L_OF, scalar

        // A loads: A[m][r] = B[node, r, i, a]; base + immediate offsets
        int rb0 = (half * 2) * 40 + k;          // chunk 0: K = 0..3
        int rb1 = (4 + half * 2) * 40 + k;      // chunk 1: K = 4..7
        float a00 = Bn[rb0], a01 = Bn[rb0 + 40];
        float a10 = Bn[rb1], a11 = Bn[rb1 + 40];

        v2f av0, av1, bv0, bv1;
        av0.x = avalid ? a00 : 0.0f;  av0.y = avalid ? a01 : 0.0f;
        av1.x = avalid ? a10 : 0.0f;  av1.y = avalid ? a11 : 0.0f;
        bv0.x = (l == 0) ? wv[0] : (l == 1) ? wv[4] : wv[8];
        bv0.y = (l == 0) ? wv[1] : (l == 1) ? wv[5] : wv[9];
        bv1.x = (l == 0) ? wv[2] : (l == 1) ? wv[6] : wv[10];
        bv1.y = (l == 0) ? wv[3] : (l == 1) ? wv[7] : wv[11];

        v8f acc = {0.f, 0.f, 0.f, 0.f, 0.f, 0.f, 0.f, 0.f};
        acc = __builtin_amdgcn_wmma_f32_16x16x4_f32(false, av0, false, bv0,
                                                    (short)0, acc, false, false);
        acc = __builtin_amdgcn_wmma_f32_16x16x4_f32(false, av1, false, bv1,
                                                    (short)0, acc, false, false);

        if (bval) {                     // only column-divergence; reconverges before next WMMA
            if (full) emit_stores<false>(out, acc, ero, n0, half, bcol, i, a, N);
            else      emit_stores<true >(out, acc, ero, n0, half, bcol, i, a, N);
        }
    }
}

extern "C" void kernel_launch(void* const* d_in, const int* in_sizes, int n_in,
                              void* d_out, int out_size, void* d_ws, size_t ws_size,
                              hipStream_t stream) {
    const int*   atomic_numbers = (const int*)d_in[0];
    const int*   edge_index     = (const int*)d_in[1];
    const float* dij            = (const float*)d_in[2];
    const float* uij            = (const float*)d_in[3];
    // d_in[4] = positions (unused)
    const float* W_send         = (const float*)d_in[5];
    const float* W_recv         = (const float*)d_in[6];
    const float* widths         = (const float*)d_in[7];
    const float* W_rad          = (const float*)d_in[8];
    float* out = (float*)d_out;

    int N = in_sizes[0];
    int E = in_sizes[2];

    // workspace: B accumulator (N*320 f32), emb_s (N*4), emb_r (N*4)
    float* B  = (float*)d_ws;
    float* es = B + (long long)N * NVALs;
    float* er = es + (long long)N * NABc;

    int nzero = N * NVALs;
    k_zero<<<(nzero + 255) / 256, 256, 0, stream>>>(B, nzero);

    k_emb<<<(N + 255) / 256, 256, 0, stream>>>(atomic_numbers, W_send, W_recv, es, er, N);

    // 8 waves (edges) per 256-thread block
    int eb = (E + 7) / 8;
    k_edge<<<eb, 256, 0, stream>>>(edge_index, dij, uij, widths, es, B, E);

    // node kernel: grid.x = node-tiles (8 per block), grid.y = 5 ia-groups
    int ntiles = (N + 15) / 16;
    dim3 ngrid((ntiles + 7) / 8, 5);
    k_node_wmma<<<ngrid, 256, 0, stream>>>(B, er, W_rad, out, N, ntiles);
}